// TransfPropModule_33397665694031
// MI455X (gfx1250) — compile-verified
//
#include <hip/hip_runtime.h>
#include <hip/hip_bf16.h>

// Edge-gather transform propagation:
//   out = [R[c_i] (N*9) | t[c_i] (N*3) | x*w + (R[c_i]*pos + t[c_i])*(1-w) (N*3)]
// Memory-bound (~0.3 FLOP/byte): optimize pure data movement.
//  - t + cluster_weights (256KB) staged into LDS via gfx1250 async global->LDS
//  - R gathered from L2 (576KB table, L2-resident)
//  - streamed outputs written with non-temporal stores

typedef float f32x4 __attribute__((ext_vector_type(4)));
typedef int   i32x4 __attribute__((ext_vector_type(4)));

#define GLOBAL_AS __attribute__((address_space(1)))
#define LDS_AS    __attribute__((address_space(3)))

__device__ __forceinline__ void async_ld16(void* lds_dst, const void* gsrc) {
#if __has_builtin(__builtin_amdgcn_global_load_async_to_lds_b128)
  __builtin_amdgcn_global_load_async_to_lds_b128(
      (GLOBAL_AS i32x4*)(const_cast<void*>(gsrc)),
      (LDS_AS    i32x4*)(lds_dst),
      /*offset=*/0, /*cpol=*/0);
#else
  *(f32x4*)lds_dst = *(const f32x4*)gsrc;   // sync fallback
#endif
}

__device__ __forceinline__ void wait_async_all() {
#if __has_builtin(__builtin_amdgcn_s_wait_asynccnt)
  __builtin_amdgcn_s_wait_asynccnt(0);
#else
  asm volatile("s_wait_asynccnt 0" ::: "memory");
#endif
}

__device__ __forceinline__ void st_nt(float* p, float v) {
  __builtin_nontemporal_store(v, p);
}

__global__ __launch_bounds__(1024, 1)
void transf_prop_kernel(const float* __restrict__ x,     // [N*3]
                        const float* __restrict__ pos,   // [N*3]
                        const float* __restrict__ cw,    // [M]
                        const float* __restrict__ R,     // [M*9]
                        const float* __restrict__ t,     // [M*3]
                        const int*   __restrict__ cidx32,// row 1 of edge_index, int64 viewed as int pairs
                        float* __restrict__ outR,        // [N*9]
                        float* __restrict__ outT,        // [N*3]
                        float* __restrict__ outX,        // [N*3]
                        int N, int M)
{
  extern __shared__ float smem[];         // M (weights) + 3*M (t) floats = 16*M bytes
  float* lds_w = smem;
  float* lds_t = smem + M;

  const int tid = threadIdx.x;
  const int nt  = blockDim.x;

  // --- cooperative staging of w and t into LDS (async, B128 granules) ---
  {
    const f32x4* wg = (const f32x4*)cw;
    f32x4*       wl = (f32x4*)lds_w;
    for (int i = tid; i < (M >> 2); i += nt)
      async_ld16(wl + i, wg + i);
    const f32x4* tg = (const f32x4*)t;
    f32x4*       tl = (f32x4*)lds_t;
    for (int i = tid; i < ((3 * M) >> 2); i += nt)
      async_ld16(tl + i, tg + i);
  }
  wait_async_all();
  __syncthreads();

  // --- grid-stride edge loop ---
  const int stride = gridDim.x * blockDim.x;
  for (int i = blockIdx.x * blockDim.x + tid; i < N; i += stride) {
    const int c = cidx32[2 * i];          // low dword of int64 cluster index (< 16384)

    const float* Rb = R + 9 * (long)c;
    const float r00 = Rb[0], r01 = Rb[1], r02 = Rb[2];
    const float r10 = Rb[3], r11 = Rb[4], r12 = Rb[5];
    const float r20 = Rb[6], r21 = Rb[7], r22 = Rb[8];

    const long i3 = 3 * (long)i;
    const float px = pos[i3 + 0], py = pos[i3 + 1], pz = pos[i3 + 2];
    const float x0 = x[i3 + 0],   x1 = x[i3 + 1],   x2 = x[i3 + 2];

    const float t0 = lds_t[3 * c + 0];
    const float t1 = lds_t[3 * c + 1];
    const float t2 = lds_t[3 * c + 2];
    const float w  = lds_w[c];

    const float xp0 = fmaf(r00, px, fmaf(r01, py, fmaf(r02, pz, t0)));
    const float xp1 = fmaf(r10, px, fmaf(r11, py, fmaf(r12, pz, t1)));
    const float xp2 = fmaf(r20, px, fmaf(r21, py, fmaf(r22, pz, t2)));

    // x*w + xp*(1-w) == xp + w*(x - xp)
    const float xo0 = fmaf(w, x0 - xp0, xp0);
    const float xo1 = fmaf(w, x1 - xp1, xp1);
    const float xo2 = fmaf(w, x2 - xp2, xp2);

    float* oR = outR + 9 * (long)i;
    st_nt(oR + 0, r00); st_nt(oR + 1, r01); st_nt(oR + 2, r02);
    st_nt(oR + 3, r10); st_nt(oR + 4, r11); st_nt(oR + 5, r12);
    st_nt(oR + 6, r20); st_nt(oR + 7, r21); st_nt(oR + 8, r22);

    float* oT = outT + i3;
    st_nt(oT + 0, t0); st_nt(oT + 1, t1); st_nt(oT + 2, t2);

    float* oX = outX + i3;
    st_nt(oX + 0, xo0); st_nt(oX + 1, xo1); st_nt(oX + 2, xo2);
  }
}

extern "C" void kernel_launch(void* const* d_in, const int* in_sizes, int n_in,
                              void* d_out, int out_size, void* d_ws, size_t ws_size,
                              hipStream_t stream) {
  const float* x   = (const float*)d_in[0];   // [N,3]
  const float* pos = (const float*)d_in[1];   // [N,3]
  // d_in[2] = batch (unused)
  const float* cw  = (const float*)d_in[3];   // [M,1]
  const float* R   = (const float*)d_in[4];   // [M,3,3]
  const float* t   = (const float*)d_in[5];   // [M,3]

  const int N = in_sizes[0] / 3;
  const int M = in_sizes[3];

  // edge_index is [2,N] int64; we need row 1 (cluster idx). View as int pairs.
  const int* cidx32 = (const int*)d_in[6] + 2 * (long long)N;

  float* out  = (float*)d_out;
  float* outR = out;                 // N*9
  float* outT = out + 9LL * N;       // N*3
  float* outX = out + 12LL * N;      // N*3

  const size_t shmem = (size_t)16 * (size_t)M;  // (M + 3M) floats = 256KB for M=16384

  dim3 block(1024);
  dim3 grid(512);                    // 8 edges/thread at N=4.19M, grid-stride handles rest

  transf_prop_kernel<<<grid, block, shmem, stream>>>(
      x, pos, cw, R, t, cidx32, outR, outT, outX, N, M);
}